// GraphNeuralNetwork_13786845020423
// MI455X (gfx1250) — compile-verified
//
#include <hip/hip_runtime.h>

#define N_NODES 10000
#define N_EDGES 320000
#define D_MODEL 256
#define D_CONT  128
#define N_LAYERS 3
#define K_OUT   4096
#define LN_EPS  1e-5f
#define ET      (N_EDGES + N_NODES)   // edges + self loops

typedef float v2f __attribute__((ext_vector_type(2)));
typedef float v8f __attribute__((ext_vector_type(8)));

// ---- init: copy x -> x_cur, zero the three int arrays -------------------
__global__ void k_init(const float* __restrict__ x, float* __restrict__ xc,
                       int* __restrict__ degi, int* __restrict__ cnt,
                       int* __restrict__ cur) {
    int i = blockIdx.x * 256 + threadIdx.x;
    if (i < N_NODES * D_MODEL) xc[i] = x[i];
    if (i < N_NODES) { degi[i] = 0; cnt[i] = 0; cur[i] = 0; }
}

// ---- count: out-degree over row (for norm), in-degree over col (for CSR)
__global__ void k_edge_pre(const int* __restrict__ row, const int* __restrict__ col,
                           int* __restrict__ degi, int* __restrict__ cnt) {
    int e = blockIdx.x * 256 + threadIdx.x;
    if (e >= ET) return;
    if (e < N_EDGES) {
        atomicAdd(&degi[row[e]], 1);
        atomicAdd(&cnt[col[e]], 1);
    } else {
        atomicAdd(&cnt[e - N_EDGES], 1);   // self-loop destination
    }
}

// ---- dis = (deg + 1_selfloop)^-1/2 --------------------------------------
__global__ void k_dis(const int* __restrict__ degi, float* __restrict__ dis) {
    int i = blockIdx.x * 256 + threadIdx.x;
    if (i < N_NODES) dis[i] = rsqrtf((float)degi[i] + 1.0f);
}

// ---- single-block exclusive prefix sum over in-degrees ------------------
#define SCAN_T  1024
#define SCAN_CH 10            // ceil(10000/1024)
__global__ __launch_bounds__(SCAN_T) void k_scan(const int* __restrict__ cnt,
                                                 int* __restrict__ rowstart) {
    __shared__ int part[SCAN_T];
    int t = threadIdx.x;
    int base = t * SCAN_CH;
    int lo[SCAN_CH];
    int s = 0;
#pragma unroll
    for (int j = 0; j < SCAN_CH; ++j) {
        int i = base + j;
        int v = (i < N_NODES) ? cnt[i] : 0;
        lo[j] = s; s += v;
    }
    part[t] = s;
    __syncthreads();
    for (int off = 1; off < SCAN_T; off <<= 1) {
        int v = (t >= off) ? part[t - off] : 0;
        __syncthreads();
        part[t] += v;
        __syncthreads();
    }
    int excl = (t == 0) ? 0 : part[t - 1];
#pragma unroll
    for (int j = 0; j < SCAN_CH; ++j) {
        int i = base + j;
        if (i < N_NODES) rowstart[i] = excl + lo[j];
    }
    if (t == SCAN_T - 1) rowstart[N_NODES] = part[SCAN_T - 1];
}

// ---- fill CSR adjacency: store SOURCE node id per (dest-ordered) slot ---
__global__ void k_fill(const int* __restrict__ row, const int* __restrict__ col,
                       const int* __restrict__ rowstart, int* __restrict__ cur,
                       int* __restrict__ adj) {
    int e = blockIdx.x * 256 + threadIdx.x;
    if (e >= ET) return;
    int r, c;
    if (e < N_EDGES) { r = row[e]; c = col[e]; }
    else             { r = e - N_EDGES; c = r; }
    int pos = atomicAdd(&cur[c], 1);
    adj[rowstart[c] + pos] = r;
}

// ---- per-node linear maps via fp32 WMMA, epilogue pre-scales by dis[row]
// ht[n, h*128+o] = dis[n] * (sum_k xc[n, h*128+k] * W_h[o,k] + b_h[o])
// One wave -> one 16x16 tile; K=128 via 32 chained V_WMMA_F32_16X16X4_F32.
__global__ __launch_bounds__(128) void k_gemm(
    const float* __restrict__ xc,
    const float* __restrict__ Wc, const float* __restrict__ bc,
    const float* __restrict__ Wp, const float* __restrict__ bp,
    const float* __restrict__ dis, float* __restrict__ ht, int layer) {
    const int lane = threadIdx.x & 31;
    const int wave = threadIdx.x >> 5;
    const int mt   = blockIdx.x;                // 0..624  (row tile)
    const int ct   = blockIdx.y * 4 + wave;     // 0..15   (col tile over D=256)
    const int half = ct >> 3;                   // 0: content, 1: position
    const int otile = (ct & 7) * 16;            // col offset within 128-half
    const float* W    = (half ? Wp : Wc) + layer * D_CONT * D_CONT;
    const float* bias = (half ? bp : bc) + layer * D_CONT;
    const int cb = half * D_CONT;

    const int lh = lane >> 4;                   // K sub-pair select
    const int lm = lane & 15;                   // M (for A) / N (for B)
    // A 16x4 f32 layout: lanes 0-15 hold K=k0,k0+1; lanes 16-31 hold K=k0+2,k0+3
    const float* Ap = xc + (mt * 16 + lm) * D_MODEL + cb + lh * 2;
    // B[k,o] = W[o,k]: symmetric layout, contiguous along k in row o
    const float* Bp = W + (otile + lm) * D_CONT + lh * 2;

    v8f acc = {0.f, 0.f, 0.f, 0.f, 0.f, 0.f, 0.f, 0.f};
#pragma unroll
    for (int k0 = 0; k0 < D_CONT; k0 += 4) {
        float2 af = *(const float2*)(Ap + k0);
        float2 bf = *(const float2*)(Bp + k0);
        v2f a; a[0] = af.x; a[1] = af.y;
        v2f b; b[0] = bf.x; b[1] = bf.y;
        acc = __builtin_amdgcn_wmma_f32_16x16x4_f32(
            false, a, false, b, (short)0, acc, false, false);
    }
    // C/D layout: VGPR v: lanes 0-15 -> M=v, N=lane; lanes 16-31 -> M=v+8
    const float bv = bias[otile + lm];
    const int orow = mt * 16 + lh * 8;
    float* Op = ht + orow * D_MODEL + cb + otile + lm;
#pragma unroll
    for (int v = 0; v < 8; ++v) Op[v * D_MODEL] = (acc[v] + bv) * dis[orow + v];
}

// ---- fused aggregation + layernorm (+relu) + residual -------------------
// One wave32 per destination node: accumulate sum of pre-scaled ht rows,
// scale by dis[c], LN via shuffle tree, residual-add into xc. Atomic-free.
__global__ __launch_bounds__(256) void k_agg_ln(
    const int* __restrict__ rowstart, const int* __restrict__ adj,
    const float* __restrict__ dis, const float* __restrict__ ht,
    const float* __restrict__ gamma, const float* __restrict__ beta,
    float* __restrict__ xc, int layer, int do_relu) {
    int lane = threadIdx.x & 31;
    int wave = threadIdx.x >> 5;
    int c = blockIdx.x * 8 + wave;
    if (c >= N_NODES) return;
    const int p0 = rowstart[c], p1 = rowstart[c + 1];
    float a0 = 0.f, a1 = 0.f, a2 = 0.f, a3 = 0.f;
    float a4 = 0.f, a5 = 0.f, a6 = 0.f, a7 = 0.f;
    int p = p0;
    // 2x unrolled: keep two independent gather chains in flight
    for (; p + 1 < p1; p += 2) {
        int r0 = adj[p];
        int r1 = adj[p + 1];
        const float* h0 = ht + r0 * D_MODEL + lane * 8;
        const float* h1 = ht + r1 * D_MODEL + lane * 8;
        float4 u0 = *(const float4*)(h0);
        float4 u1 = *(const float4*)(h0 + 4);
        float4 w0 = *(const float4*)(h1);
        float4 w1 = *(const float4*)(h1 + 4);
        a0 += u0.x + w0.x; a1 += u0.y + w0.y; a2 += u0.z + w0.z; a3 += u0.w + w0.w;
        a4 += u1.x + w1.x; a5 += u1.y + w1.y; a6 += u1.z + w1.z; a7 += u1.w + w1.w;
    }
    if (p < p1) {
        int r = adj[p];
        const float* hp = ht + r * D_MODEL + lane * 8;
        float4 u0 = *(const float4*)(hp);
        float4 u1 = *(const float4*)(hp + 4);
        a0 += u0.x; a1 += u0.y; a2 += u0.z; a3 += u0.w;
        a4 += u1.x; a5 += u1.y; a6 += u1.z; a7 += u1.w;
    }
    const float wc = dis[c];
    float hv[8] = {a0 * wc, a1 * wc, a2 * wc, a3 * wc,
                   a4 * wc, a5 * wc, a6 * wc, a7 * wc};
    // LayerNorm over the 256-wide row held across the wave
    float s = 0.f, q = 0.f;
#pragma unroll
    for (int j = 0; j < 8; ++j) { s += hv[j]; q += hv[j] * hv[j]; }
#pragma unroll
    for (int off = 16; off > 0; off >>= 1) {
        s += __shfl_xor(s, off, 32);
        q += __shfl_xor(q, off, 32);
    }
    const float mu  = s * (1.0f / D_MODEL);
    const float var = q * (1.0f / D_MODEL) - mu * mu;
    const float rs  = rsqrtf(var + LN_EPS);
    const float* g = gamma + layer * D_MODEL + lane * 8;
    const float* b = beta  + layer * D_MODEL + lane * 8;
    float* xp = xc + c * D_MODEL + lane * 8;
#pragma unroll
    for (int j = 0; j < 8; ++j) {
        float y = (hv[j] - mu) * rs * g[j] + b[j];
        if (do_relu) y = fmaxf(y, 0.0f);
        xp[j] += y;
    }
}

// ---- gather nodes_of_interest -------------------------------------------
__global__ void k_gather(const int* __restrict__ noi, const float* __restrict__ xc,
                         float* __restrict__ out) {
    int i = blockIdx.x * 256 + threadIdx.x;
    if (i < K_OUT * D_MODEL) {
        int k = i >> 8;      // D_MODEL == 256
        int d = i & 255;
        out[i] = xc[noi[k] * D_MODEL + d];
    }
}

extern "C" void kernel_launch(void* const* d_in, const int* in_sizes, int n_in,
                              void* d_out, int out_size, void* d_ws, size_t ws_size,
                              hipStream_t stream) {
    (void)in_sizes; (void)n_in; (void)out_size; (void)ws_size;
    const float* x   = (const float*)d_in[0];
    const int*   ei  = (const int*)d_in[1];          // (2, E) flat
    const int*   noi = (const int*)d_in[2];
    const float* Wc  = (const float*)d_in[3];
    const float* bc  = (const float*)d_in[4];
    const float* Wp  = (const float*)d_in[5];
    const float* bp  = (const float*)d_in[6];
    const float* gm  = (const float*)d_in[7];
    const float* bt  = (const float*)d_in[8];
    float* out = (float*)d_out;

    const int* row = ei;
    const int* col = ei + N_EDGES;

    const size_t ND = (size_t)N_NODES * D_MODEL;
    float* ws  = (float*)d_ws;
    float* xc  = ws;                  // N*D
    float* ht  = ws + ND;             // N*D  (pre-scaled by dis[row])
    float* dis = ws + 2 * ND;         // N
    int* ib       = (int*)(ws + 2 * ND + N_NODES);
    int* degi     = ib;                       // N
    int* cnt      = ib + N_NODES;             // N
    int* cur      = ib + 2 * N_NODES;         // N
    int* rowstart = ib + 3 * N_NODES;         // N+1
    int* adj      = ib + 4 * N_NODES + 1;     // ET

    const int BLK = 256;
    const int gND  = (int)((ND + BLK - 1) / BLK);            // 10000
    const int gN   = (N_NODES + BLK - 1) / BLK;              // 40
    const int gET  = (ET + BLK - 1) / BLK;                   // 1290
    const int gWv  = (N_NODES + 7) / 8;                      // 1250
    const int gOut = (K_OUT * D_MODEL + BLK - 1) / BLK;      // 4096

    // --- build normalization + dest-ordered CSR (structure is layer-invariant)
    k_init<<<gND, BLK, 0, stream>>>(x, xc, degi, cnt, cur);
    k_edge_pre<<<gET, BLK, 0, stream>>>(row, col, degi, cnt);
    k_dis<<<gN, BLK, 0, stream>>>(degi, dis);
    k_scan<<<1, SCAN_T, 0, stream>>>(cnt, rowstart);
    k_fill<<<gET, BLK, 0, stream>>>(row, col, rowstart, cur, adj);

    // --- layers: WMMA linear (scaled epilogue) -> fused gather+LN+residual
    for (int l = 0; l < N_LAYERS; ++l) {
        k_gemm<<<dim3(N_NODES / 16, 4), 128, 0, stream>>>(xc, Wc, bc, Wp, bp, dis, ht, l);
        k_agg_ln<<<gWv, BLK, 0, stream>>>(rowstart, adj, dis, ht, gm, bt, xc, l,
                                          (l < N_LAYERS - 1) ? 1 : 0);
    }
    k_gather<<<gOut, BLK, 0, stream>>>(noi, xc, out);
}